// BottomPool_68710886801777
// MI455X (gfx1250) — compile-verified
//
#include <hip/hip_runtime.h>

// BottomPool: out[n,c,h,w] = max over h' in [0, h] of in[n,c,h',w]
// Shape (16, 256, 128, 128) fp32. Pure streaming scan along H:
// 512 MiB total traffic -> ~22 us at 23.3 TB/s HBM. Bandwidth-bound.
//
// Mapping: one thread owns one float4 column (4 consecutive w) of one
// (n,c) slice and serially walks h=0..127 with a running max in VGPRs.
// A wave32 covers exactly one 512-byte row per iteration (32 lanes x 16B),
// so every global_load_b128 / global_store_b128 is a contiguous 512B burst.
// Non-temporal hints keep the 2x256MiB streams from thrashing the 192MB L2;
// global_prefetch_b8 pulls the column 16 rows (8KB) ahead.

typedef __attribute__((ext_vector_type(4))) float v4f;

__global__ __launch_bounds__(256) void BottomPool_kernel(
    const float* __restrict__ in, float* __restrict__ out, int nCols4) {
  constexpr int H = 128;
  constexpr int W = 128;
  constexpr int W4 = W / 4;  // 32 float4 columns per row

  int idx = blockIdx.x * blockDim.x + threadIdx.x;
  if (idx >= nCols4) return;

  int slice = idx / W4;        // n*C + c   (compile-time shift by 5)
  int col4  = idx % W4;        // which float4 within the row

  size_t base = (size_t)slice * (H * W) + (size_t)col4 * 4;
  const v4f* __restrict__ ip = (const v4f*)(in + base);
  v4f* __restrict__ op = (v4f*)(out + base);

  v4f m;
  m.x = -__builtin_inff();
  m.y = -__builtin_inff();
  m.z = -__builtin_inff();
  m.w = -__builtin_inff();

#pragma unroll 4
  for (int h = 0; h < H; ++h) {
    if ((h & 3) == 0) {
      // Prefetch 16 rows (8KB) ahead; speculative OOB prefetches at the
      // tail are silently dropped by hardware.
      __builtin_prefetch((const void*)(ip + (size_t)(h + 16) * W4), 0, 0);
    }
    v4f x = __builtin_nontemporal_load(ip + (size_t)h * W4);
    m.x = fmaxf(m.x, x.x);
    m.y = fmaxf(m.y, x.y);
    m.z = fmaxf(m.z, x.z);
    m.w = fmaxf(m.w, x.w);
    __builtin_nontemporal_store(m, op + (size_t)h * W4);
  }
}

extern "C" void kernel_launch(void* const* d_in, const int* in_sizes, int n_in,
                              void* d_out, int out_size, void* d_ws, size_t ws_size,
                              hipStream_t stream) {
  (void)n_in; (void)d_ws; (void)ws_size; (void)out_size;
  const float* in = (const float*)d_in[0];
  float* out = (float*)d_out;

  constexpr int H = 128;
  constexpr int W = 128;
  int total = in_sizes[0];                 // 16*256*128*128
  int slices = total / (H * W);            // N*C = 4096
  int nCols4 = slices * (W / 4);           // 131072 float4 columns

  int block = 256;                         // 8 wave32 per workgroup
  int grid = (nCols4 + block - 1) / block; // 512 blocks
  BottomPool_kernel<<<grid, block, 0, stream>>>(in, out, nCols4);
}